// GAT_Node_Classifier_49761491091780
// MI455X (gfx1250) — compile-verified
//
#include <hip/hip_runtime.h>
#include <math.h>

#define N_NODES 50000
#define N_EDGES 800000
#define IN_DIM  256
#define HID     32
#define HEADS   8
#define OUT_DIM 16
#define F1      (HEADS * HID)   // 256

typedef __attribute__((ext_vector_type(2))) float v2f;
typedef __attribute__((ext_vector_type(8))) float v8f;

// ---------------------------------------------------------------------------
// Sign-aware float atomic max via integer atomics (exact, handles -0 / -inf).
// For non-negative floats the int ordering matches float ordering;
// for negatives the unsigned ordering is reversed.
// ---------------------------------------------------------------------------
__device__ __forceinline__ void atomic_max_f32(float* addr, float val) {
    int iv = __float_as_int(val);
    if (iv >= 0) {
        atomicMax((int*)addr, iv);
    } else {
        atomicMin((unsigned int*)addr, (unsigned int)iv);
    }
}

// ---------------------------------------------------------------------------
// Init: zero accumulators, set segment-max buffers to -inf.
// ---------------------------------------------------------------------------
__global__ void init_kernel(float* __restrict__ agg1, float* __restrict__ m1,
                            float* __restrict__ dn1, float* __restrict__ m2,
                            float* __restrict__ dn2, float* __restrict__ agg2) {
    int tid = blockIdx.x * blockDim.x + threadIdx.x;
    if (tid < N_NODES * F1)      agg1[tid] = 0.0f;
    if (tid < N_NODES * HEADS)   { m1[tid] = -INFINITY; dn1[tid] = 0.0f; }
    if (tid < N_NODES)           { m2[tid] = -INFINITY; dn2[tid] = 0.0f; }
    if (tid < N_NODES * OUT_DIM) agg2[tid] = 0.0f;
}

// ---------------------------------------------------------------------------
// GEMM1: h[50000,256] = emb[50000,256] @ W1[256,256]   (f32 WMMA 16x16x4)
// Block = 256 threads = 8 waves. Block owns 16 rows; wave w owns cols [16w,16w+16).
// A tile (16x256) staged in LDS with stride 260 (bank-conflict-free frags).
// ---------------------------------------------------------------------------
__global__ __launch_bounds__(256) void gemm1_kernel(const float* __restrict__ emb,
                                                    const float* __restrict__ W1,
                                                    float* __restrict__ h) {
    __shared__ float As[16 * 260];
    const int m0   = blockIdx.x * 16;
    const int t    = threadIdx.x;

    // cooperative, fully coalesced load of the 16x256 A tile
    #pragma unroll
    for (int i = 0; i < 16; ++i)
        As[i * 260 + t] = emb[m0 * IN_DIM + i * IN_DIM + t];
    __syncthreads();

    const int wave = t >> 5;
    const int lane = t & 31;
    const int half = lane >> 4;   // 0: K+{0,1}, 1: K+{2,3}
    const int lm   = lane & 15;
    const int n0   = wave * 16;

    v8f acc = {};
    #pragma unroll 4
    for (int k = 0; k < IN_DIM; k += 4) {
        const int ka = k + 2 * half;
        v2f a = *(const v2f*)&As[lm * 260 + ka];            // A[lm][ka], A[lm][ka+1]
        v2f b;
        b.x = W1[ka * F1 + n0 + lm];                        // B[ka][n0+lm]
        b.y = W1[(ka + 1) * F1 + n0 + lm];                  // B[ka+1][n0+lm]
        acc = __builtin_amdgcn_wmma_f32_16x16x4_f32(false, a, false, b,
                                                    (short)0, acc, false, false);
    }
    #pragma unroll
    for (int r = 0; r < 8; ++r)
        h[(m0 + r + 8 * half) * F1 + n0 + lm] = acc[r];
}

// ---------------------------------------------------------------------------
// GEMM2: g[50000,16] = h2[50000,256] @ W2[256,16]   (f32 WMMA 16x16x4)
// One wave per 16-row tile (output is only 16 columns wide).
// ---------------------------------------------------------------------------
__global__ __launch_bounds__(256) void gemm2_kernel(const float* __restrict__ h2,
                                                    const float* __restrict__ W2,
                                                    float* __restrict__ g) {
    const int wid  = (blockIdx.x * blockDim.x + threadIdx.x) >> 5;
    const int lane = threadIdx.x & 31;
    const int m0   = wid * 16;
    if (m0 >= N_NODES) return;                // wave-uniform: EXEC all-ones for WMMA
    const int half = lane >> 4;
    const int lm   = lane & 15;

    v8f acc = {};
    #pragma unroll 4
    for (int k = 0; k < F1; k += 4) {
        const int ka = k + 2 * half;
        v2f a = *(const v2f*)&h2[(m0 + lm) * F1 + ka];
        v2f b;
        b.x = W2[ka * OUT_DIM + lm];
        b.y = W2[(ka + 1) * OUT_DIM + lm];
        acc = __builtin_amdgcn_wmma_f32_16x16x4_f32(false, a, false, b,
                                                    (short)0, acc, false, false);
    }
    #pragma unroll
    for (int r = 0; r < 8; ++r)
        g[(m0 + r + 8 * half) * OUT_DIM + lm] = acc[r];
}

// ---------------------------------------------------------------------------
// Per-node attention logits, layer 1: el/er [N, 8]
// ---------------------------------------------------------------------------
__global__ void elr1_kernel(const float* __restrict__ h, const float* __restrict__ al,
                            const float* __restrict__ ar, float* __restrict__ el,
                            float* __restrict__ er) {
    int tid = blockIdx.x * blockDim.x + threadIdx.x;
    if (tid >= N_NODES * HEADS) return;
    const int n = tid >> 3, hh = tid & 7;
    const float* hp  = h  + n * F1 + hh * HID;
    const float* alp = al + hh * HID;
    const float* arp = ar + hh * HID;
    float sl = 0.0f, sr = 0.0f;
    #pragma unroll
    for (int d = 0; d < HID; ++d) { float v = hp[d]; sl += v * alp[d]; sr += v * arp[d]; }
    el[tid] = sl;
    er[tid] = sr;
}

// ---------------------------------------------------------------------------
// Edge scores + segment max, layer 1
// ---------------------------------------------------------------------------
__global__ void scores1_kernel(const int* __restrict__ src, const int* __restrict__ dst,
                               const float* __restrict__ el, const float* __restrict__ er,
                               float* __restrict__ e1, float* __restrict__ m1) {
    int tid = blockIdx.x * blockDim.x + threadIdx.x;
    if (tid >= N_EDGES * HEADS) return;
    const int e = tid >> 3, hh = tid & 7;
    float s = el[src[e] * HEADS + hh] + er[dst[e] * HEADS + hh];
    s = (s > 0.0f) ? s : 0.2f * s;            // LeakyReLU(0.2)
    e1[tid] = s;
    atomic_max_f32(&m1[dst[e] * HEADS + hh], s);
}

__global__ void expsum1_kernel(const int* __restrict__ dst, float* __restrict__ e1,
                               const float* __restrict__ m1, float* __restrict__ dn1) {
    int tid = blockIdx.x * blockDim.x + threadIdx.x;
    if (tid >= N_EDGES * HEADS) return;
    const int e = tid >> 3, hh = tid & 7;
    const int d = dst[e];
    float t = expf(e1[tid] - m1[d * HEADS + hh]);
    e1[tid] = t;
    atomicAdd(&dn1[d * HEADS + hh], t);
}

// ---------------------------------------------------------------------------
// Weighted scatter-sum, layer 1. One wave per edge; lane l handles feature
// (32*head + l) -> fully coalesced 128B loads and 128B atomic bursts.
// ---------------------------------------------------------------------------
__global__ __launch_bounds__(256) void agg1_kernel(const int* __restrict__ src,
                                                   const int* __restrict__ dst,
                                                   const float* __restrict__ h,
                                                   const float* __restrict__ ee,
                                                   const float* __restrict__ dn,
                                                   float* __restrict__ agg) {
    const int wid  = (blockIdx.x * blockDim.x + threadIdx.x) >> 5;
    const int lane = threadIdx.x & 31;
    if (wid >= N_EDGES) return;
    const int s = src[wid], d = dst[wid];
    const float* hs = h   + s * F1;
    float*       ad = agg + d * F1;
    #pragma unroll
    for (int j = 0; j < HEADS; ++j) {
        const float alpha = ee[wid * HEADS + j] / dn[d * HEADS + j];
        atomicAdd(&ad[j * 32 + lane], hs[j * 32 + lane] * alpha);
    }
}

// ---------------------------------------------------------------------------
// h2 = ELU(agg1 + bias1)  (written over h; h is rebuilt by gemm1 each call)
// ---------------------------------------------------------------------------
__global__ void bias_elu_kernel(const float* __restrict__ agg, const float* __restrict__ bias,
                                float* __restrict__ h2) {
    int tid = blockIdx.x * blockDim.x + threadIdx.x;
    if (tid >= N_NODES * F1) return;
    float v = agg[tid] + bias[tid & (F1 - 1)];
    h2[tid] = (v > 0.0f) ? v : (expf(v) - 1.0f);
}

// ---------------------------------------------------------------------------
// Layer 2 (single head, 16-dim)
// ---------------------------------------------------------------------------
__global__ void elr2_kernel(const float* __restrict__ g, const float* __restrict__ al,
                            const float* __restrict__ ar, float* __restrict__ el,
                            float* __restrict__ er) {
    int n = blockIdx.x * blockDim.x + threadIdx.x;
    if (n >= N_NODES) return;
    float sl = 0.0f, sr = 0.0f;
    #pragma unroll
    for (int d = 0; d < OUT_DIM; ++d) { float v = g[n * OUT_DIM + d]; sl += v * al[d]; sr += v * ar[d]; }
    el[n] = sl;
    er[n] = sr;
}

__global__ void scores2_kernel(const int* __restrict__ src, const int* __restrict__ dst,
                               const float* __restrict__ el, const float* __restrict__ er,
                               float* __restrict__ e2, float* __restrict__ m2) {
    int e = blockIdx.x * blockDim.x + threadIdx.x;
    if (e >= N_EDGES) return;
    float s = el[src[e]] + er[dst[e]];
    s = (s > 0.0f) ? s : 0.2f * s;
    e2[e] = s;
    atomic_max_f32(&m2[dst[e]], s);
}

__global__ void expsum2_kernel(const int* __restrict__ dst, float* __restrict__ e2,
                               const float* __restrict__ m2, float* __restrict__ dn2) {
    int e = blockIdx.x * blockDim.x + threadIdx.x;
    if (e >= N_EDGES) return;
    const int d = dst[e];
    float t = expf(e2[e] - m2[d]);
    e2[e] = t;
    atomicAdd(&dn2[d], t);
}

__global__ void agg2_kernel(const int* __restrict__ src, const int* __restrict__ dst,
                            const float* __restrict__ g, const float* __restrict__ ee,
                            const float* __restrict__ dn, float* __restrict__ agg) {
    int tid = blockIdx.x * blockDim.x + threadIdx.x;
    if (tid >= N_EDGES * OUT_DIM) return;
    const int e = tid >> 4, d = tid & 15;
    const int dn_node = dst[e];
    const float alpha = ee[e] / dn[dn_node];
    atomicAdd(&agg[dn_node * OUT_DIM + d], g[src[e] * OUT_DIM + d] * alpha);
}

__global__ void final_kernel(const float* __restrict__ agg, const float* __restrict__ bias,
                             float* __restrict__ out) {
    int tid = blockIdx.x * blockDim.x + threadIdx.x;
    if (tid >= N_NODES * OUT_DIM) return;
    out[tid] = agg[tid] + bias[tid & (OUT_DIM - 1)];
}

// ---------------------------------------------------------------------------
extern "C" void kernel_launch(void* const* d_in, const int* in_sizes, int n_in,
                              void* d_out, int out_size, void* d_ws, size_t ws_size,
                              hipStream_t stream) {
    const float* emb = (const float*)d_in[0];
    const int*   src = (const int*)  d_in[1];
    const int*   dst = (const int*)  d_in[2];
    const float* W1  = (const float*)d_in[3];
    const float* al1 = (const float*)d_in[4];
    const float* ar1 = (const float*)d_in[5];
    const float* b1  = (const float*)d_in[6];
    const float* W2  = (const float*)d_in[7];
    const float* al2 = (const float*)d_in[8];
    const float* ar2 = (const float*)d_in[9];
    const float* b2  = (const float*)d_in[10];
    float* out = (float*)d_out;

    char* p = (char*)d_ws;
    auto alloc = [&](size_t elems) {
        float* r = (float*)p;
        p += ((elems * sizeof(float)) + 255) & ~(size_t)255;
        return r;
    };
    float* h    = alloc((size_t)N_NODES * F1);        // h, later reused as h2
    float* agg1 = alloc((size_t)N_NODES * F1);
    float* e1   = alloc((size_t)N_EDGES * HEADS);
    float* el1  = alloc((size_t)N_NODES * HEADS);
    float* er1  = alloc((size_t)N_NODES * HEADS);
    float* m1   = alloc((size_t)N_NODES * HEADS);
    float* dn1  = alloc((size_t)N_NODES * HEADS);
    float* g    = alloc((size_t)N_NODES * OUT_DIM);
    float* e2   = alloc((size_t)N_EDGES);
    float* el2  = alloc((size_t)N_NODES);
    float* er2  = alloc((size_t)N_NODES);
    float* m2   = alloc((size_t)N_NODES);
    float* dn2  = alloc((size_t)N_NODES);
    float* agg2 = alloc((size_t)N_NODES * OUT_DIM);

    const int B = 256;
    init_kernel<<<(N_NODES * F1 + B - 1) / B, B, 0, stream>>>(agg1, m1, dn1, m2, dn2, agg2);

    // ----- layer 1 -----
    gemm1_kernel<<<N_NODES / 16, B, 0, stream>>>(emb, W1, h);
    elr1_kernel<<<(N_NODES * HEADS + B - 1) / B, B, 0, stream>>>(h, al1, ar1, el1, er1);
    scores1_kernel<<<(N_EDGES * HEADS + B - 1) / B, B, 0, stream>>>(src, dst, el1, er1, e1, m1);
    expsum1_kernel<<<(N_EDGES * HEADS + B - 1) / B, B, 0, stream>>>(dst, e1, m1, dn1);
    agg1_kernel<<<(N_EDGES * 32 + B - 1) / B, B, 0, stream>>>(src, dst, h, e1, dn1, agg1);
    bias_elu_kernel<<<(N_NODES * F1 + B - 1) / B, B, 0, stream>>>(agg1, b1, h);  // h2 in h

    // ----- layer 2 -----
    {
        int waves = N_NODES / 16;
        gemm2_kernel<<<(waves * 32 + B - 1) / B, B, 0, stream>>>(h, W2, g);
    }
    elr2_kernel<<<(N_NODES + B - 1) / B, B, 0, stream>>>(g, al2, ar2, el2, er2);
    scores2_kernel<<<(N_EDGES + B - 1) / B, B, 0, stream>>>(src, dst, el2, er2, e2, m2);
    expsum2_kernel<<<(N_EDGES + B - 1) / B, B, 0, stream>>>(dst, e2, m2, dn2);
    agg2_kernel<<<(N_EDGES * OUT_DIM + B - 1) / B, B, 0, stream>>>(src, dst, g, e2, dn2, agg2);
    final_kernel<<<(N_NODES * OUT_DIM + B - 1) / B, B, 0, stream>>>(agg2, b2, out);
}